// MNDGCNConv_17806934409758
// MI455X (gfx1250) — compile-verified
//
#include <hip/hip_runtime.h>
#include <hip/hip_bf16.h>

#define NN 100000
#define TT 3
#define EE 1600000
#define DD 128
#define ALPHA_C 0.5f
#define BN_EPS_C 1e-5f
#define TAU_EFF 1.1f   // exp(0)+0.1

typedef float v2f __attribute__((ext_vector_type(2)));
typedef float v8f __attribute__((ext_vector_type(8)));

// ---------------- l1 row-normalize: x1 = x / max(sum|x|,1e-12) ----------------
__global__ void l1norm_kernel(const float* __restrict__ x, float* __restrict__ x1) {
  int wave = blockIdx.x * (blockDim.x >> 5) + (threadIdx.x >> 5);
  int lane = threadIdx.x & 31;
  if (wave >= NN) return;
  const float4 a = ((const float4*)(x + (size_t)wave * DD))[lane];
  float s = fabsf(a.x) + fabsf(a.y) + fabsf(a.z) + fabsf(a.w);
#pragma unroll
  for (int off = 16; off; off >>= 1) s += __shfl_xor(s, off, 32);
  float inv = 1.0f / fmaxf(s, 1e-12f);
  float4 o; o.x = a.x * inv; o.y = a.y * inv; o.z = a.z * inv; o.w = a.w * inv;
  ((float4*)(x1 + (size_t)wave * DD))[lane] = o;
}

// ---------------- per-edge degree counting ----------------
__global__ void degree_kernel(const int* __restrict__ row, const int* __restrict__ col,
                              float* __restrict__ outdeg, float* __restrict__ indeg) {
  int e = blockIdx.x * blockDim.x + threadIdx.x;
  if (e >= EE) return;
  atomicAdd(&outdeg[row[e]], 1.0f);
  atomicAdd(&indeg[col[e]], 1.0f);
}

__global__ void invsqrt_kernel(float* __restrict__ d, int n) {
  int i = blockIdx.x * blockDim.x + threadIdx.x;
  if (i >= n) return;
  float v = d[i];
  d[i] = (v > 0.0f) ? rsqrtf(v) : 0.0f;
}

__device__ __forceinline__ void atom_add4(float* p, float v0, float v1, float v2, float v3) {
  atomicAdd(p + 0, v0); atomicAdd(p + 1, v1); atomicAdd(p + 2, v2); atomicAdd(p + 3, v3);
}

// ---------------- fused 6-way SpMM: one wave per edge, 4 feats per lane --------
__global__ void spmm_kernel(const int* __restrict__ row, const int* __restrict__ col,
                            const float* __restrict__ invout, const float* __restrict__ invin,
                            const float* __restrict__ x, const float* __restrict__ x1,
                            float* __restrict__ out_nei, float* __restrict__ in_nei,
                            float* __restrict__ s_osq, float* __restrict__ s_o1,
                            float* __restrict__ s_isq, float* __restrict__ s_i1) {
  int wave = blockIdx.x * (blockDim.x >> 5) + (threadIdx.x >> 5);
  int lane = threadIdx.x & 31;
  if (wave >= EE) return;
  int r = row[wave], c = col[wave];
  float v = invout[r] * invin[c];           // == val == val_t
  size_t rb = (size_t)r * DD + lane * 4;
  size_t cb = (size_t)c * DD + lane * 4;
  // scatter to row: gather at col
  float4 xc  = *(const float4*)(x  + cb);
  float4 nc  = *(const float4*)(x1 + cb);
  atom_add4(out_nei + rb, v * xc.x, v * xc.y, v * xc.z, v * xc.w);
  atom_add4(s_o1    + rb, v * nc.x, v * nc.y, v * nc.z, v * nc.w);
  atom_add4(s_osq   + rb, v * nc.x * nc.x, v * nc.y * nc.y, v * nc.z * nc.z, v * nc.w * nc.w);
  // scatter to col: gather at row
  float4 xr  = *(const float4*)(x  + rb);
  float4 nr  = *(const float4*)(x1 + rb);
  atom_add4(in_nei + cb, v * xr.x, v * xr.y, v * xr.z, v * xr.w);
  atom_add4(s_i1   + cb, v * nr.x, v * nr.y, v * nr.z, v * nr.w);
  atom_add4(s_isq  + cb, v * nr.x * nr.x, v * nr.y * nr.y, v * nr.z * nr.z, v * nr.w * nr.w);
}

// ---------------- per-node gating coefficients (one wave per node) -------------
__global__ void coeff_kernel(const float* __restrict__ x1,
                             const float* __restrict__ s_osq, const float* __restrict__ s_o1,
                             const float* __restrict__ s_isq, const float* __restrict__ s_i1,
                             const int* __restrict__ odeg_idx, const int* __restrict__ ideg_idx,
                             const float* __restrict__ oemb, const float* __restrict__ iemb,
                             const float* __restrict__ W_outf, const float* __restrict__ b_outf,
                             const float* __restrict__ W_inf, const float* __restrict__ b_inf,
                             const float* __restrict__ omask, const float* __restrict__ omaskb,
                             const float* __restrict__ imask, const float* __restrict__ imaskb,
                             float* __restrict__ Cout_t, float* __restrict__ Cin_t,
                             float* __restrict__ Cout_sum, float* __restrict__ Cin_sum) {
  int n = blockIdx.x * (blockDim.x >> 5) + (threadIdx.x >> 5);
  int lane = threadIdx.x & 31;
  if (n >= NN) return;
  size_t b = (size_t)n * DD + lane * 4;
  float4 xv  = *(const float4*)(x1 + b);
  float4 aoq = *(const float4*)(s_osq + b);
  float4 ao1 = *(const float4*)(s_o1 + b);
  float4 aiq = *(const float4*)(s_isq + b);
  float4 ai1 = *(const float4*)(s_i1 + b);
  float diag = (n == NN - 1) ? 2.0f : 1.0f;  // missing last diagonal quirk
  int oi = odeg_idx[n], ii = ideg_idx[n];
  float4 oe = *(const float4*)(oemb + (size_t)oi * DD + lane * 4);
  float4 ie = *(const float4*)(iemb + (size_t)ii * DD + lane * 4);
  float4 wo = *(const float4*)(W_outf + lane * 4);
  float4 wi = *(const float4*)(W_inf + lane * 4);
  float co = 0.0f, ci = 0.0f;
#define ACC(f) { \
    float xs = xv.f * xv.f; \
    float eo = diag * xs + aoq.f - 2.0f * ao1.f * xv.f; \
    float ei = diag * xs + aiq.f - 2.0f * ai1.f * xv.f; \
    co += wo.f * (oe.f - eo); \
    ci += wi.f * (ie.f - ei); }
  ACC(x) ACC(y) ACC(z) ACC(w)
#undef ACC
#pragma unroll
  for (int off = 16; off; off >>= 1) { co += __shfl_xor(co, off, 32); ci += __shfl_xor(ci, off, 32); }
  co += b_outf[0]; ci += b_inf[0];
  float c0 = 1.0f / (1.0f + __expf((ci - co) * (1.0f / TAU_EFF)));  // softmax([co,ci])[0]
  float Co = c0 * omask[n] + omaskb[n];
  float Ci = (1.0f - c0) * imask[n] + imaskb[n];
  if (lane == 0) {
    Cout_t[n] = Co; Cin_t[n] = Ci;
    Cout_sum[n] += Co; Cin_sum[n] += Ci;
  }
}

// -------- fused dual WMMA GEMM: acc += Cout*(A_src@Wsrc+bsrc) + Cin*(A_dst@Wdst+bdst)
__global__ __launch_bounds__(256)
void gemm_combine_kernel(const float* __restrict__ Asrc, const float* __restrict__ Adst,
                         const float* __restrict__ Wsrc, const float* __restrict__ bsrc,
                         const float* __restrict__ Wdst, const float* __restrict__ bdst,
                         const float* __restrict__ Cout, const float* __restrict__ Cin,
                         float* __restrict__ acc) {
  __shared__ float As[16][132];
  __shared__ float Ad[16][132];
  int tid = threadIdx.x;
  int nodebase = blockIdx.x * 16;
  {
    int r = tid >> 4, cc = (tid & 15) * 8;
    const float4* s0 = (const float4*)(Asrc + (size_t)(nodebase + r) * DD + cc);
    const float4* d0 = (const float4*)(Adst + (size_t)(nodebase + r) * DD + cc);
    float4 a0 = s0[0], a1 = s0[1], b0 = d0[0], b1 = d0[1];
    *(float4*)&As[r][cc] = a0; *(float4*)&As[r][cc + 4] = a1;
    *(float4*)&Ad[r][cc] = b0; *(float4*)&Ad[r][cc + 4] = b1;
  }
  __syncthreads();
  int lane = tid & 31, half = lane >> 4, ncol = lane & 15, mrow = lane & 15;
  int colbase = (tid >> 5) * 16;
  float bs = bsrc[colbase + ncol], bd = bdst[colbase + ncol];
  v8f cs, cd;
#pragma unroll
  for (int j = 0; j < 8; ++j) { cs[j] = bs; cd[j] = bd; }
#pragma unroll 4
  for (int k = 0; k < DD; k += 4) {
    int ka = k + 2 * half;
    v2f a, b;
    a[0] = As[mrow][ka]; a[1] = As[mrow][ka + 1];
    b[0] = Wsrc[(size_t)ka * DD + colbase + ncol];
    b[1] = Wsrc[(size_t)(ka + 1) * DD + colbase + ncol];
    cs = __builtin_amdgcn_wmma_f32_16x16x4_f32(false, a, false, b, (short)0, cs, false, false);
    a[0] = Ad[mrow][ka]; a[1] = Ad[mrow][ka + 1];
    b[0] = Wdst[(size_t)ka * DD + colbase + ncol];
    b[1] = Wdst[(size_t)(ka + 1) * DD + colbase + ncol];
    cd = __builtin_amdgcn_wmma_f32_16x16x4_f32(false, a, false, b, (short)0, cd, false, false);
  }
#pragma unroll
  for (int j = 0; j < 8; ++j) {
    int rowg = nodebase + j + 8 * half;
    size_t idx = (size_t)rowg * DD + colbase + ncol;
    acc[idx] += Cout[rowg] * cs[j] + Cin[rowg] * cd[j];
  }
}

// -------- final WMMA GEMM: pre = acc/3 + 0.5*(x@Wfc + bfc), in place ----------
__global__ __launch_bounds__(256)
void gemm_fc_kernel(const float* __restrict__ X, const float* __restrict__ Wfc,
                    const float* __restrict__ bfc, float* __restrict__ acc) {
  __shared__ float As[16][132];
  int tid = threadIdx.x;
  int nodebase = blockIdx.x * 16;
  {
    int r = tid >> 4, cc = (tid & 15) * 8;
    const float4* s0 = (const float4*)(X + (size_t)(nodebase + r) * DD + cc);
    float4 a0 = s0[0], a1 = s0[1];
    *(float4*)&As[r][cc] = a0; *(float4*)&As[r][cc + 4] = a1;
  }
  __syncthreads();
  int lane = tid & 31, half = lane >> 4, ncol = lane & 15, mrow = lane & 15;
  int colbase = (tid >> 5) * 16;
  float bv = bfc[colbase + ncol];
  v8f c;
#pragma unroll
  for (int j = 0; j < 8; ++j) c[j] = bv;
#pragma unroll 4
  for (int k = 0; k < DD; k += 4) {
    int ka = k + 2 * half;
    v2f a, b;
    a[0] = As[mrow][ka]; a[1] = As[mrow][ka + 1];
    b[0] = Wfc[(size_t)ka * DD + colbase + ncol];
    b[1] = Wfc[(size_t)(ka + 1) * DD + colbase + ncol];
    c = __builtin_amdgcn_wmma_f32_16x16x4_f32(false, a, false, b, (short)0, c, false, false);
  }
#pragma unroll
  for (int j = 0; j < 8; ++j) {
    int rowg = nodebase + j + 8 * half;
    size_t idx = (size_t)rowg * DD + colbase + ncol;
    acc[idx] = acc[idx] * (1.0f / 3.0f) + ALPHA_C * c[j];
  }
}

// ---------------- batchnorm statistics / apply ----------------
__global__ void bn_stats_kernel(const float* __restrict__ pre,
                                float* __restrict__ sum, float* __restrict__ sumsq) {
  int col = threadIdx.x & 127;
  int sub = threadIdx.x >> 7;
  float s = 0.0f, ss = 0.0f;
  for (int r = blockIdx.x * 2 + sub; r < NN; r += gridDim.x * 2) {
    float v = pre[(size_t)r * DD + col];
    s += v; ss += v * v;
  }
  atomicAdd(&sum[col], s);
  atomicAdd(&sumsq[col], ss);
}

__global__ void bn_apply_kernel(const float* __restrict__ pre,
                                const float* __restrict__ sum, const float* __restrict__ sumsq,
                                const float* __restrict__ gamma, const float* __restrict__ beta,
                                float* __restrict__ out) {
  size_t i = (size_t)blockIdx.x * blockDim.x + threadIdx.x;
  if (i >= (size_t)NN * DD) return;
  int col = (int)(i & 127);
  float m = sum[col] * (1.0f / NN);
  float var = sumsq[col] * (1.0f / NN) - m * m;
  out[i] = (pre[i] - m) * rsqrtf(var + BN_EPS_C) * gamma[col] + beta[col];
}

__global__ void cmean_kernel(const float* __restrict__ cin_sum, const float* __restrict__ cout_sum,
                             float* __restrict__ out) {
  int n = blockIdx.x * blockDim.x + threadIdx.x;
  if (n >= NN) return;
  out[(size_t)NN * DD + n]      = cin_sum[n]  * (1.0f / 3.0f);
  out[(size_t)NN * DD + NN + n] = cout_sum[n] * (1.0f / 3.0f);
}

extern "C" void kernel_launch(void* const* d_in, const int* in_sizes, int n_in,
                              void* d_out, int out_size, void* d_ws, size_t ws_size,
                              hipStream_t stream) {
  (void)in_sizes; (void)n_in; (void)out_size; (void)ws_size;
  const float* x            = (const float*)d_in[0];
  const int*   edge_index   = (const int*)d_in[1];
  const int*   in_deg_idx   = (const int*)d_in[2];
  const int*   out_deg_idx  = (const int*)d_in[3];
  const float* out_mask     = (const float*)d_in[4];
  const float* out_mask_b   = (const float*)d_in[5];
  const float* in_mask      = (const float*)d_in[6];
  const float* in_mask_b    = (const float*)d_in[7];
  const float* W_src        = (const float*)d_in[8];
  const float* b_src        = (const float*)d_in[9];
  const float* W_dst        = (const float*)d_in[10];
  const float* b_dst        = (const float*)d_in[11];
  const float* W_outf       = (const float*)d_in[12];
  const float* b_outf       = (const float*)d_in[13];
  const float* W_inf        = (const float*)d_in[14];
  const float* b_inf        = (const float*)d_in[15];
  const float* W_fc         = (const float*)d_in[16];
  const float* b_fc         = (const float*)d_in[17];
  const float* in_deg_emb   = (const float*)d_in[18];
  const float* out_deg_emb  = (const float*)d_in[19];
  const float* gamma        = (const float*)d_in[20];
  const float* beta         = (const float*)d_in[21];

  const size_t ND = (size_t)NN * DD;
  float* ws       = (float*)d_ws;
  float* x1       = ws;
  float* out_nei  = x1 + ND;       // out_nei..s_i1 contiguous (6*ND) for one memset
  float* in_nei   = out_nei + ND;
  float* s_osq    = in_nei + ND;
  float* s_o1     = s_osq + ND;
  float* s_isq    = s_o1 + ND;
  float* s_i1     = s_isq + ND;
  float* acc      = s_i1 + ND;
  float* degs     = acc + ND;               // [2*NN]: outdeg then indeg
  float* Cout_t   = degs + 2 * NN;
  float* Cin_t    = Cout_t + NN;
  float* Cout_sum = Cin_t + NN;             // Cout_sum..Cin_sum contiguous
  float* Cin_sum  = Cout_sum + NN;
  float* bnsum    = Cin_sum + NN;           // [128] + [128]
  float* bnsumsq  = bnsum + DD;

  hipMemsetAsync(acc, 0, ND * sizeof(float), stream);
  hipMemsetAsync(Cout_sum, 0, 2 * NN * sizeof(float), stream);
  hipMemsetAsync(bnsum, 0, 2 * DD * sizeof(float), stream);

  l1norm_kernel<<<(NN + 7) / 8, 256, 0, stream>>>(x, x1);

  for (int t = 0; t < TT; ++t) {
    const int* row = edge_index + (size_t)t * 2 * EE;
    const int* col = row + EE;
    hipMemsetAsync(degs, 0, 2 * NN * sizeof(float), stream);
    hipMemsetAsync(out_nei, 0, 6 * ND * sizeof(float), stream);
    degree_kernel<<<(EE + 255) / 256, 256, 0, stream>>>(row, col, degs, degs + NN);
    invsqrt_kernel<<<(2 * NN + 255) / 256, 256, 0, stream>>>(degs, 2 * NN);
    spmm_kernel<<<(EE + 7) / 8, 256, 0, stream>>>(row, col, degs, degs + NN, x, x1,
                                                  out_nei, in_nei, s_osq, s_o1, s_isq, s_i1);
    coeff_kernel<<<(NN + 7) / 8, 256, 0, stream>>>(x1, s_osq, s_o1, s_isq, s_i1,
        out_deg_idx + (size_t)t * NN, in_deg_idx + (size_t)t * NN,
        out_deg_emb, in_deg_emb, W_outf, b_outf, W_inf, b_inf,
        out_mask + (size_t)t * NN, out_mask_b + (size_t)t * NN,
        in_mask + (size_t)t * NN, in_mask_b + (size_t)t * NN,
        Cout_t, Cin_t, Cout_sum, Cin_sum);
    gemm_combine_kernel<<<NN / 16, 256, 0, stream>>>(out_nei, in_nei, W_src, b_src,
                                                     W_dst, b_dst, Cout_t, Cin_t, acc);
  }

  gemm_fc_kernel<<<NN / 16, 256, 0, stream>>>(x, W_fc, b_fc, acc);
  bn_stats_kernel<<<512, 256, 0, stream>>>(acc, bnsum, bnsumsq);
  bn_apply_kernel<<<(int)((ND + 255) / 256), 256, 0, stream>>>(acc, bnsum, bnsumsq,
                                                               gamma, beta, (float*)d_out);
  cmean_kernel<<<(NN + 255) / 256, 256, 0, stream>>>(Cin_sum, Cout_sum, (float*)d_out);
}